// Proposal_5531917877589
// MI455X (gfx1250) — compile-verified
//
#include <hip/hip_runtime.h>
#include <stdint.h>

// ---------------------------------------------------------------------------
// RPN Proposal (decode + greedy NMS, k=2000) for MI455X (gfx1250, wave32).
//
// Roofline reasoning:
//  * decode pass: 200k * ~36B = 7.2MB -> ~0.3us @ 23.3 TB/s. Irrelevant.
//  * NMS: 2000 strictly-serial selections, each = global argmax(200k) +
//    suppress(200k). FLOPs (~6 GFLOP) and bytes (4.8MB working set, fully
//    L2-resident in 192MB) are both trivial; the cost is 2000 device-wide
//    synchronizations. So: one persistent kernel, ONE global barrier per
//    selection, all per-iteration data in LDS (staged once with CDNA5
//    global_load_async_to_lds_b128), only an 8B argmax key + 20B winner
//    box crossing blocks via L2 atomics.
// ---------------------------------------------------------------------------

#define S_TOT   200000
#define NB      100                 // persistent blocks (all resident)
#define NT      256                 // 8 wave32 waves per block
#define SLICE   (S_TOT / NB)        // 2000 elements per block (exact)
#define KSEL    2000                // boxes selected
#define NMS_TH  0.8f
#define NEGV    (-1e30f)
#define EPSV    (1e-8f)
#define MINSZ   16.0f

// Monotonic float->uint map so unsigned compare == float compare.
__device__ __forceinline__ unsigned mono_f32(float f) {
  unsigned u = __float_as_uint(f);
  return (u & 0x80000000u) ? ~u : (u | 0x80000000u);
}

// CDNA5 async copy: 16B per active lane, global -> LDS. Tracked by ASYNCcnt.
__device__ __forceinline__ void async_b128_to_lds(void* lds_dst, const void* gsrc) {
  unsigned           loff  = (unsigned)(uintptr_t)lds_dst;       // LDS byte offset
  unsigned long long gaddr = (unsigned long long)(uintptr_t)gsrc;
  asm volatile("global_load_async_to_lds_b128 %0, %1, off"
               :: "v"(loff), "v"(gaddr) : "memory");
}
__device__ __forceinline__ void wait_asynccnt0() {
  asm volatile("s_wait_asynccnt 0x0" ::: "memory");
}

// ---------------------------------------------------------------------------
// Kernel 1: box decode + clip + min-size filter -> SoA workspace.
// Also (re)initializes the NMS sync area every call (graph-replay safe).
// ---------------------------------------------------------------------------
__global__ __launch_bounds__(NT) void proposal_decode(
    const float* __restrict__ cls,   // (S,1)
    const float* __restrict__ reg,   // (S,4)
    const float* __restrict__ anc,   // (S,4)
    const float* __restrict__ img,   // (2,) = {H, W}
    float* __restrict__ gx1, float* __restrict__ gy1,
    float* __restrict__ gx2, float* __restrict__ gy2,
    float* __restrict__ gar, float* __restrict__ gsc,
    unsigned long long* __restrict__ sync)   // [0]=counter, [1..KSEL]=argmax keys
{
  int i = blockIdx.x * NT + threadIdx.x;
  if (i <= KSEL) sync[i] = 0ULL;           // fresh barrier/argmax state per launch
  if (i >= S_TOT) return;

  float H = img[0], W = img[1];
  float ax1 = anc[4*i+0], ay1 = anc[4*i+1], ax2 = anc[4*i+2], ay2 = anc[4*i+3];
  float dx  = reg[4*i+0], dy  = reg[4*i+1], dw  = reg[4*i+2], dh  = reg[4*i+3];

  float wa  = ax2 - ax1 + 1.0f;
  float ha  = ay2 - ay1 + 1.0f;
  float cxa = ax1 + 0.5f * wa;
  float cya = ay1 + 0.5f * ha;
  float cx  = dx * wa + cxa;
  float cy  = dy * ha + cya;
  float w   = expf(dw) * wa;
  float h   = expf(dh) * ha;

  float x1 = fminf(fmaxf(cx - 0.5f * w, 0.0f), W);
  float y1 = fminf(fmaxf(cy - 0.5f * h, 0.0f), H);
  float x2 = fminf(fmaxf(cx + 0.5f * w, 0.0f), W);
  float y2 = fminf(fmaxf(cy + 0.5f * h, 0.0f), H);

  float ws_ = x2 - x1 + 1.0f;
  float hs_ = y2 - y1 + 1.0f;
  bool keep = (ws_ >= MINSZ) && (hs_ >= MINSZ);

  gx1[i] = x1; gy1[i] = y1; gx2[i] = x2; gy2[i] = y2;
  gar[i] = (x2 - x1) * (y2 - y1);
  gsc[i] = keep ? cls[i] : NEGV;
}

// ---------------------------------------------------------------------------
// Kernel 2: persistent greedy NMS. 100 blocks, each owns a 2000-element
// slice staged into LDS via async-to-LDS. One monotonic-counter grid
// barrier per selection; fused suppress(prev winner) + argmax pass.
// ---------------------------------------------------------------------------
__global__ __launch_bounds__(NT) void proposal_nms(
    const float* __restrict__ gx1, const float* __restrict__ gy1,
    const float* __restrict__ gx2, const float* __restrict__ gy2,
    const float* __restrict__ gar, const float* __restrict__ gsc,
    unsigned long long* __restrict__ sync,   // [0]=counter, [1+t]=key slot t
    float* __restrict__ out)                 // (KSEL,4)
{
  __shared__ __align__(16) float sx1[SLICE], sy1[SLICE], sx2[SLICE], sy2[SLICE];
  __shared__ __align__(16) float sar[SLICE], ssc[SLICE];
  __shared__ unsigned long long sred[NT / 32];
  __shared__ float sbc[5];       // winner x1,y1,x2,y2,area broadcast
  __shared__ unsigned swin;      // winner global index broadcast

  const int tid   = threadIdx.x;
  const int gbase = blockIdx.x * SLICE;
  unsigned long long* counter = &sync[0];
  unsigned long long* gkey    = &sync[1];

  // ---- stage this block's slice into LDS (async DMA path, 16B per lane) ----
  for (int i = tid * 4; i < SLICE; i += NT * 4) {
    async_b128_to_lds(&sx1[i], gx1 + gbase + i);
    async_b128_to_lds(&sy1[i], gy1 + gbase + i);
    async_b128_to_lds(&sx2[i], gx2 + gbase + i);
    async_b128_to_lds(&sy2[i], gy2 + gbase + i);
    async_b128_to_lds(&sar[i], gar + gbase + i);
    async_b128_to_lds(&ssc[i], gsc + gbase + i);
  }
  wait_asynccnt0();     // each wave waits for its own async copies
  __syncthreads();      // then all waves' copies are visible

  float wx1 = 0.f, wy1 = 0.f, wx2 = 0.f, wy2 = 0.f, war = 0.f;
  unsigned pwin = 0xFFFFFFFFu;

  for (int t = 0; t < KSEL; ++t) {
    // ---- fused: suppress vs winner(t-1), then local argmax ----
    unsigned long long best = 0ULL;
    for (int i = tid; i < SLICE; i += NT) {
      float sc = ssc[i];
      if (t > 0) {
        float ix1 = fmaxf(wx1, sx1[i]);
        float iy1 = fmaxf(wy1, sy1[i]);
        float ix2 = fminf(wx2, sx2[i]);
        float iy2 = fminf(wy2, sy2[i]);
        float inter = fmaxf(ix2 - ix1, 0.0f) * fmaxf(iy2 - iy1, 0.0f);
        float iou   = inter / (war + sar[i] - inter + EPSV);
        unsigned gidx = (unsigned)(gbase + i);
        if (iou > NMS_TH || gidx == pwin) { sc = NEGV; ssc[i] = NEGV; }
      }
      // higher score wins; on ties, smaller index wins (matches jnp.argmax)
      unsigned long long key =
          ((unsigned long long)mono_f32(sc) << 32) |
          (unsigned long long)(0xFFFFFFFFu - (unsigned)(gbase + i));
      if (key > best) best = key;
    }

    // ---- wave32 reduction, then cross-wave via LDS ----
    for (int o = 16; o > 0; o >>= 1) {
      unsigned long long other = __shfl_xor(best, o, 32);
      if (other > best) best = other;
    }
    if ((tid & 31) == 0) sred[tid >> 5] = best;
    __syncthreads();

    if (tid == 0) {
      unsigned long long b = sred[0];
      #pragma unroll
      for (int wv = 1; wv < NT / 32; ++wv)
        if (sred[wv] > b) b = sred[wv];
      atomicMax(&gkey[t], b);                 // fresh slot per iteration: no reset race
      __threadfence();
      atomicAdd(counter, 1ULL);               // monotonic grid barrier
      unsigned long long target = (unsigned long long)NB * (unsigned long long)(t + 1);
      while (__hip_atomic_load(counter, __ATOMIC_RELAXED, __HIP_MEMORY_SCOPE_AGENT) < target)
        __builtin_amdgcn_s_sleep(1);
      __threadfence();
      unsigned long long kwin =
          __hip_atomic_load(&gkey[t], __ATOMIC_RELAXED, __HIP_MEMORY_SCOPE_AGENT);
      unsigned widx = 0xFFFFFFFFu - (unsigned)(kwin & 0xFFFFFFFFull);
      swin   = widx;
      float bx1 = gx1[widx], by1 = gy1[widx];   // L2-hot, read-only
      float bx2 = gx2[widx], by2 = gy2[widx];
      sbc[0] = bx1; sbc[1] = by1; sbc[2] = bx2; sbc[3] = by2; sbc[4] = gar[widx];
      if (blockIdx.x == 0) {                    // emit selection t
        out[4 * t + 0] = bx1; out[4 * t + 1] = by1;
        out[4 * t + 2] = bx2; out[4 * t + 3] = by2;
      }
    }
    __syncthreads();
    wx1 = sbc[0]; wy1 = sbc[1]; wx2 = sbc[2]; wy2 = sbc[3]; war = sbc[4];
    pwin = swin;
  }
}

// ---------------------------------------------------------------------------
extern "C" void kernel_launch(void* const* d_in, const int* in_sizes, int n_in,
                              void* d_out, int out_size, void* d_ws, size_t ws_size,
                              hipStream_t stream) {
  const float* cls = (const float*)d_in[0];   // (1,S,1)
  const float* reg = (const float*)d_in[1];   // (1,S,4)
  const float* anc = (const float*)d_in[2];   // (1,S,4)
  const float* img = (const float*)d_in[3];   // (2,)

  float* f   = (float*)d_ws;
  float* gx1 = f + 0 * S_TOT;
  float* gy1 = f + 1 * S_TOT;
  float* gx2 = f + 2 * S_TOT;
  float* gy2 = f + 3 * S_TOT;
  float* gar = f + 4 * S_TOT;
  float* gsc = f + 5 * S_TOT;
  unsigned long long* sync = (unsigned long long*)(f + 6 * S_TOT);  // KSEL+1 slots

  int nblk = (S_TOT + NT - 1) / NT;
  proposal_decode<<<nblk, NT, 0, stream>>>(cls, reg, anc, img,
                                           gx1, gy1, gx2, gy2, gar, gsc, sync);
  proposal_nms<<<NB, NT, 0, stream>>>(gx1, gy1, gx2, gy2, gar, gsc,
                                      sync, (float*)d_out);
  (void)in_sizes; (void)n_in; (void)out_size; (void)ws_size;
}